// PositionalEncoding_80659485819003
// MI455X (gfx1250) — compile-verified
//
#include <hip/hip_runtime.h>

// out[b,s,:] = x[b,s,:] + (positions[b,s] >= 0 ? pe[positions[b,s], :] : 0)
//
// Memory-bound gather+add (382 MB stream @ 23.3 TB/s => ~16 us floor).
// MI455X strategy:
//  - one row per wave32: 32 lanes x 4 float4 chunks, fully unrolled
//    (D=512 hardcoded fast path) for 4 B128 loads in flight per wave
//  - non-temporal hints on x/out (single-use stream) so the 192 MB L2 stays
//    dedicated to the 747 KB pe table (gather always hits cache)
//  - readfirstlane(positions[row]) -> SGPR -> scalar per-wave branch on the
//    mask; pe addressing on the SALU, no per-lane predication
//  - x loads hoisted above the branch: shared by both mask paths, max MLP

typedef float v4f __attribute__((ext_vector_type(4)));

// ---------------------------------------------------------------------------
// Fast path: D = 512 (defined first so its asm is inspectable up front).
// ---------------------------------------------------------------------------
__global__ __launch_bounds__(256) void pe_add_d512(
    const float* __restrict__ x,
    const int*   __restrict__ positions,
    const float* __restrict__ pe,
    float*       __restrict__ out,
    int n_rows)
{
    constexpr int D      = 512;
    constexpr int CHUNKS = D / (32 * 4);            // 4 float4 chunks per lane

    const int wave = threadIdx.x >> 5;              // 8 waves per block
    const int lane = threadIdx.x & 31;
    const int row  = blockIdx.x * 8 + wave;         // one row per wave
    if (row >= n_rows) return;

    // Wave-uniform position -> SGPR -> scalar branch.
    const int pos = __builtin_amdgcn_readfirstlane(positions[row]);

    const float* xr   = x   + (size_t)row * D + lane * 4;
    float*       orow = out + (size_t)row * D + lane * 4;

    // Issue all x loads up front (non-temporal, streaming).
    v4f v[CHUNKS];
#pragma unroll
    for (int c = 0; c < CHUNKS; ++c)
        v[c] = __builtin_nontemporal_load((const v4f*)(xr + c * 128));

    if (pos >= 0) {
        const float* pr = pe + (size_t)pos * D + lane * 4;
#pragma unroll
        for (int c = 0; c < CHUNKS; ++c) {
            // pe is hot/shared: regular temporal load (L2/WGP$ resident).
            v4f pv = *(const v4f*)(pr + c * 128);
            v[c] += pv;
        }
    }

#pragma unroll
    for (int c = 0; c < CHUNKS; ++c)
        __builtin_nontemporal_store(v[c], (v4f*)(orow + c * 128));
}

// ---------------------------------------------------------------------------
// Generic fallback for any other D.
// ---------------------------------------------------------------------------
__global__ __launch_bounds__(256) void pe_add_generic(
    const float* __restrict__ x,
    const int*   __restrict__ positions,
    const float* __restrict__ pe,
    float*       __restrict__ out,
    int n_rows, int D)
{
    const int wave = threadIdx.x >> 5;
    const int lane = threadIdx.x & 31;
    const int row  = blockIdx.x * 8 + wave;
    if (row >= n_rows) return;

    const int pos = __builtin_amdgcn_readfirstlane(positions[row]);
    const size_t xbase = (size_t)row * (size_t)D;

    if (pos >= 0) {
        const size_t pbase = (size_t)pos * (size_t)D;
        int i = lane * 4;
        for (; i + 3 < D; i += 32 * 4) {
            v4f xv = __builtin_nontemporal_load((const v4f*)(x + xbase + i));
            v4f pv = *(const v4f*)(pe + pbase + i);
            xv += pv;
            __builtin_nontemporal_store(xv, (v4f*)(out + xbase + i));
        }
        for (int j = i; j < D; ++j)  // tail (D % 4 != 0)
            out[xbase + j] = x[xbase + j] + pe[pbase + j];
    } else {
        int i = lane * 4;
        for (; i + 3 < D; i += 32 * 4) {
            v4f xv = __builtin_nontemporal_load((const v4f*)(x + xbase + i));
            __builtin_nontemporal_store(xv, (v4f*)(out + xbase + i));
        }
        for (int j = i; j < D; ++j)
            out[xbase + j] = x[xbase + j];
    }
}

extern "C" void kernel_launch(void* const* d_in, const int* in_sizes, int n_in,
                              void* d_out, int out_size, void* d_ws, size_t ws_size,
                              hipStream_t stream)
{
    (void)n_in; (void)out_size; (void)d_ws; (void)ws_size;

    const float* x         = (const float*)d_in[0];  // [B,S,D] fp32
    const int*   positions = (const int*)  d_in[1];  // [B,S]   int32
    const float* pe        = (const float*)d_in[2];  // [L,D]   fp32
    float*       out       = (float*)d_out;          // [B,S,D] fp32

    const int n_rows = in_sizes[1];                  // B*S = 93440
    const int D      = in_sizes[0] / in_sizes[1];    // 512

    const int rows_per_block = 8;                    // 8 waves x 32 lanes
    const int blocks = (n_rows + rows_per_block - 1) / rows_per_block;

    if (D == 512) {
        pe_add_d512<<<dim3(blocks), dim3(256), 0, stream>>>(
            x, positions, pe, out, n_rows);
    } else {
        pe_add_generic<<<dim3(blocks), dim3(256), 0, stream>>>(
            x, positions, pe, out, n_rows, D);
    }
}